// Non_local_6786048328412
// MI455X (gfx1250) — compile-verified
//
#include <hip/hip_runtime.h>

typedef __attribute__((ext_vector_type(2))) float v2f;
typedef __attribute__((ext_vector_type(8))) float v8f;

#define BATCH 8
#define CCH 256
#define NSP (48*48)                 // 2304 spatial positions
#define TILES_PER_B (NSP/16)        // 144
#define NTILES (BATCH*TILES_PER_B)  // 1152 wave-tiles
#define WAVES_PER_BLK 8
#define EPSV 1e-5f

// Workspace layout (floats):
//  [0, 3456)            per-wave partials: {sum(phi*g), sum(theta), sum(theta^2)}
//  [3456, 21888)        theta[b*N+n]
//  [21888, 40320)       yv[b*N+n] = theta*S_b - ybar
//  [40320, 40576)       s[c]
//  [40576, 40584)       S_b
//  [40584, 40585)       ybar
#define WS_PART  0
#define WS_THETA 3456
#define WS_YV    21888
#define WS_S     40320
#define WS_SB    40576
#define WS_YBAR  40584

// ---------------------------------------------------------------------------
// Kernel 1: fused g/theta/phi projections via V_WMMA_F32_16X16X4_F32.
// One wave handles one (batch, 16-column) tile; K-loop over C in steps of 4.
// ---------------------------------------------------------------------------
__global__ __launch_bounds__(256) void k_proj(
    const float* __restrict__ x,
    const float* __restrict__ g_w,  const float* __restrict__ g_b,
    const float* __restrict__ th_w, const float* __restrict__ th_b,
    const float* __restrict__ ph_w, const float* __restrict__ ph_b,
    float* __restrict__ ws_theta,   float* __restrict__ ws_part)
{
    __shared__ float ldsW[4][CCH];   // rows: 0=g, 1=theta, 2=phi, 3=zeros
    const int t = threadIdx.x;
    ldsW[0][t] = g_w[t];
    ldsW[1][t] = th_w[t];
    ldsW[2][t] = ph_w[t];
    ldsW[3][t] = 0.0f;
    __syncthreads();

    const int wave = blockIdx.x * WAVES_PER_BLK + (t >> 5);
    const int lane = t & 31;
    const int b    = wave / TILES_PER_B;
    const int tile = wave - b * TILES_PER_B;
    const int m    = lane & 15;            // A-row / B-column within tile
    const int half = lane >> 4;            // K sub-pair selector
    const int mc   = (m < 3) ? m : 3;      // weight rows >=3 read zeros
    const int col  = tile * 16 + m;        // global spatial index (lanes 0-15)

    const float* __restrict__ xb = x + (size_t)b * CCH * NSP + col;

    v8f acc = {};
    #pragma unroll 4
    for (int c = 0; c < CCH; c += 4) {
        const int k = c + 2 * half;
        v2f a;  a.x = ldsW[mc][k];                 a.y = ldsW[mc][k + 1];
        v2f bm; bm.x = xb[(size_t)k * NSP];        bm.y = xb[(size_t)(k + 1) * NSP];
        acc = __builtin_amdgcn_wmma_f32_16x16x4_f32(
            /*neg_a=*/false, a, /*neg_b=*/false, bm,
            /*c_mod=*/(short)0, acc, /*reuse_a=*/false, /*reuse_b=*/false);
    }

    // Lanes 0-15 hold accumulator rows M=0..7: row0=g, row1=theta, row2=phi.
    const float gv = acc[0] + g_b[0];
    const float tv = acc[1] + th_b[0];
    const float pv = acc[2] + ph_b[0];

    const bool active = (lane < 16);
    float prod = active ? pv * gv : 0.0f;
    float s1   = active ? tv      : 0.0f;
    float s2   = active ? tv * tv : 0.0f;
    if (active) ws_theta[(size_t)b * NSP + col] = tv;

    // reduce within the 16 active lanes (width-16 sub-groups of the wave32)
    #pragma unroll
    for (int off = 8; off > 0; off >>= 1) {
        prod += __shfl_down(prod, off, 16);
        s1   += __shfl_down(s1,   off, 16);
        s2   += __shfl_down(s2,   off, 16);
    }
    if (lane == 0) {
        ws_part[wave * 3 + 0] = prod;
        ws_part[wave * 3 + 1] = s1;
        ws_part[wave * 3 + 2] = s2;
    }
}

// ---------------------------------------------------------------------------
// Kernel 2: fold per-wave partials into S_b, ybar, var_y, and per-channel s[c].
// ---------------------------------------------------------------------------
__global__ __launch_bounds__(256) void k_scalars(
    const float* __restrict__ ws_part,
    const float* __restrict__ w_w,
    const float* __restrict__ bn_gamma,
    float* __restrict__ ws_s, float* __restrict__ ws_Sb, float* __restrict__ ws_ybar)
{
    __shared__ float sS[BATCH], sT1[BATCH], sT2[BATCH];
    __shared__ float sVar;
    const int t = threadIdx.x;
    if (t < BATCH) {
        float P = 0.0f, T1 = 0.0f, T2 = 0.0f;
        for (int w = 0; w < TILES_PER_B; ++w) {
            const float* p = ws_part + (size_t)(t * TILES_PER_B + w) * 3;
            P += p[0]; T1 += p[1]; T2 += p[2];
        }
        const float S = P / (float)NSP;      // S_b = (1/N) * sum(phi*g)
        sS[t] = S; sT1[t] = T1; sT2[t] = T2;
        ws_Sb[t] = S;
    }
    __syncthreads();
    if (t == 0) {
        float sy = 0.0f, sy2 = 0.0f;
        for (int b = 0; b < BATCH; ++b) {
            sy  += sS[b] * sT1[b];
            sy2 += sS[b] * sS[b] * sT2[b];
        }
        const float inv  = 1.0f / (float)(BATCH * NSP);
        const float ybar = sy * inv;
        sVar = sy2 * inv - ybar * ybar;
        ws_ybar[0] = ybar;
    }
    __syncthreads();
    const float ww = w_w[t];
    ws_s[t] = bn_gamma[t] * ww * rsqrtf(ww * ww * sVar + EPSV);
}

// ---------------------------------------------------------------------------
// Kernel 3: yv[b,n] = theta[b,n]*S_b - ybar   (B*N = 18432 elements)
// ---------------------------------------------------------------------------
__global__ __launch_bounds__(256) void k_yv(
    const float* __restrict__ ws_theta, const float* __restrict__ ws_Sb,
    const float* __restrict__ ws_ybar,  float* __restrict__ ws_yv)
{
    const int i = blockIdx.x * 256 + threadIdx.x;
    const int b = i / NSP;
    ws_yv[i] = ws_theta[i] * ws_Sb[b] - ws_ybar[0];
}

// ---------------------------------------------------------------------------
// Kernel 4: out = x + s[c]*yv[b,n] + beta[c], float4 streaming.
// ---------------------------------------------------------------------------
__global__ __launch_bounds__(256) void k_final(
    const float* __restrict__ x,   const float* __restrict__ ws_yv,
    const float* __restrict__ ws_s, const float* __restrict__ bn_beta,
    float* __restrict__ out)
{
    const size_t i = (size_t)blockIdx.x * 256 + threadIdx.x;  // float4 index
    const size_t e = i * 4;
    const int n = (int)(e % NSP);
    const int c = (int)((e / NSP) % CCH);
    const int b = (int)(e / ((size_t)NSP * CCH));

    const float4 xv = ((const float4*)x)[i];
    const float4 yv = *(const float4*)(ws_yv + (size_t)b * NSP + n);
    const float s  = ws_s[c];
    const float be = bn_beta[c];

    float4 o;
    o.x = fmaf(s, yv.x, xv.x) + be;
    o.y = fmaf(s, yv.y, xv.y) + be;
    o.z = fmaf(s, yv.z, xv.z) + be;
    o.w = fmaf(s, yv.w, xv.w) + be;
    ((float4*)out)[i] = o;
}

// ---------------------------------------------------------------------------
extern "C" void kernel_launch(void* const* d_in, const int* in_sizes, int n_in,
                              void* d_out, int out_size, void* d_ws, size_t ws_size,
                              hipStream_t stream) {
    const float* x        = (const float*)d_in[0];
    const float* g_w      = (const float*)d_in[1];
    const float* g_b      = (const float*)d_in[2];
    const float* theta_w  = (const float*)d_in[3];
    const float* theta_b  = (const float*)d_in[4];
    const float* phi_w    = (const float*)d_in[5];
    const float* phi_b    = (const float*)d_in[6];
    const float* w_w      = (const float*)d_in[7];
    // d_in[8] = w_b: cancels inside training-mode BatchNorm, unused.
    const float* bn_gamma = (const float*)d_in[9];
    const float* bn_beta  = (const float*)d_in[10];
    float* out = (float*)d_out;
    float* ws  = (float*)d_ws;

    float* ws_part  = ws + WS_PART;
    float* ws_theta = ws + WS_THETA;
    float* ws_yv    = ws + WS_YV;
    float* ws_s     = ws + WS_S;
    float* ws_Sb    = ws + WS_SB;
    float* ws_ybar  = ws + WS_YBAR;

    // 1) WMMA projections + per-wave partial sums
    k_proj<<<NTILES / WAVES_PER_BLK, 256, 0, stream>>>(
        x, g_w, g_b, theta_w, theta_b, phi_w, phi_b, ws_theta, ws_part);

    // 2) scalar statistics + per-channel scale
    k_scalars<<<1, 256, 0, stream>>>(ws_part, w_w, bn_gamma, ws_s, ws_Sb, ws_ybar);

    // 3) yv = theta*S_b - ybar
    k_yv<<<(BATCH * NSP) / 256, 256, 0, stream>>>(ws_theta, ws_Sb, ws_ybar, ws_yv);

    // 4) residual + BN-affine stream (x should hit L2: 18.9MB << 192MB)
    k_final<<<(BATCH * CCH * NSP) / (4 * 256), 256, 0, stream>>>(
        x, ws_yv, ws_s, bn_beta, out);
}